// GraphBlock_32916629357099
// MI455X (gfx1250) — compile-verified
//
#include <hip/hip_runtime.h>
#include <cstdint>

typedef __attribute__((ext_vector_type(2))) float v2f;
typedef __attribute__((ext_vector_type(8))) float v8f;

#define NB 4
#define NN 512
#define DD 64
#define JT 32      // j-tile staged in LDS (double buffered)
#define NTILE 8    // n-rows per block in kernel 2

// -----------------------------------------------------------------------------
// Kernel 1: fused projection  [ a+b_msg | c ] = V @ W_msg^T  via fp32 WMMA.
// One wave per 16x16 output tile; K=64 consumed in 16 steps of
// V_WMMA_F32_16X16X4_F32 (full fp32 precision, matching the reference).
// -----------------------------------------------------------------------------
__global__ __launch_bounds__(32)
void proj_wmma_kernel(const float* __restrict__ V,
                      const float* __restrict__ Wmsg,
                      const float* __restrict__ bmsg,
                      float* __restrict__ aplus,   // (B,N,64) = a + b_msg
                      float* __restrict__ cbuf)    // (B,N,64) = c
{
    const int tile    = blockIdx.x;        // 4 * 32 * 8 = 1024 tiles
    const int b       = tile >> 8;
    const int rem     = tile & 255;
    const int rowtile = rem >> 3;
    const int coltile = rem & 7;
    const int n0 = rowtile << 4;
    const int m0 = coltile << 4;

    const int l    = threadIdx.x;
    const int half = l >> 4;               // lanes 16..31 hold K=+2,+3
    const int lr   = l & 15;

    const float* __restrict__ Vrow = V + (size_t)(b * NN + n0 + lr) * DD;
    const float* __restrict__ Wrow = Wmsg + (size_t)(m0 + lr) * 128;

    v8f acc = {};
#pragma unroll
    for (int k0 = 0; k0 < DD; k0 += 4) {
        const int ko = k0 + 2 * half;
        v2f af, bf;
        af.x = Vrow[ko + 0];
        af.y = Vrow[ko + 1];
        bf.x = Wrow[ko + 0];               // B[k][m] = W_msg[m][k]
        bf.y = Wrow[ko + 1];
        acc = __builtin_amdgcn_wmma_f32_16x16x4_f32(
            false, af, false, bf, (short)0, acc, false, false);
    }

    const bool  is_a = (coltile < 4);
    const float bias = is_a ? bmsg[m0 + lr] : 0.0f;
    float* __restrict__ dst = is_a ? aplus : cbuf;
    const int col = is_a ? (m0 + lr) : (m0 - 64 + lr);
#pragma unroll
    for (int r = 0; r < 8; ++r) {
        const int row = n0 + r + 8 * half;
        dst[(size_t)(b * NN + row) * DD + col] = acc[r] + bias;
    }
}

// -----------------------------------------------------------------------------
// Kernel 2: fused masked relu-aggregation + node MLP.
// 256 threads = 8 n-rows x 32 lanes, 2 features per thread.  The c tiles are
// staged by GLOBAL_LOAD_ASYNC_TO_LDS_B128 into a double buffer so the copy of
// tile t+1 overlaps the FMA loop of tile t (ASYNCcnt-tracked, barrier-fenced).
// -----------------------------------------------------------------------------
__global__ __launch_bounds__(256)
void agg_mlp_kernel(const float* __restrict__ V,
                    const unsigned char* __restrict__ mask,  // (B,N) bool
                    const float* __restrict__ aplus,
                    const float* __restrict__ cbuf,
                    const float* __restrict__ Wnode,         // (64,128)
                    const float* __restrict__ bnode,         // (64)
                    float* __restrict__ out)                 // (B,N,64)
{
    __shared__ float cs[2][JT * DD];     // 2 x 8 KB double-buffered c tiles
    __shared__ float mf[JT];             // mask as 0/1 float
    __shared__ float WnS[DD * 129];      // padded W_node (33 KB, conflict-free)
    __shared__ float aggS[NTILE * DD];
    __shared__ float vS[NTILE * DD];

    const int t  = threadIdx.x;
    const int mp = (t & 31) * 2;                // feature pair
    const int nl = t >> 5;                      // 0..7 local row
    const int b     = blockIdx.x >> 6;          // 64 chunks per batch
    const int chunk = blockIdx.x & 63;
    const int n = chunk * NTILE + nl;

    for (int i = t; i < DD * 128; i += 256)
        WnS[(i >> 7) * 129 + (i & 127)] = Wnode[i];

    const size_t bn = (size_t)(b * NN + n);
    const float2 av = *(const float2*)(aplus + bn * DD + mp);
    float accx = 0.0f, accy = 0.0f;

    const float* __restrict__ cB = cbuf + (size_t)b * NN * DD;
    const unsigned char* __restrict__ mB = mask + b * NN;
    const uint64_t cbase = (uint64_t)(uintptr_t)cB;

    // Low 32 bits of a generic shared pointer == LDS byte address (ISA 10.2).
    const uint32_t lds0 = (uint32_t)(uintptr_t)&cs[0][0];
    const uint32_t lds1 = (uint32_t)(uintptr_t)&cs[1][0];
    const uint32_t lchunk = (uint32_t)(t * 16);   // 16B per lane per issue

    // Prologue: async-copy tile 0 into buffer 0 (2 x b128 per lane = 8 KB/WG).
    asm volatile(
        "global_load_async_to_lds_b128 %0, %2, %4\n\t"
        "global_load_async_to_lds_b128 %1, %3, %4"
        :: "v"(lds0 + lchunk), "v"(lds0 + lchunk + 4096),
           "v"(lchunk), "v"(lchunk + 4096), "s"(cbase)
        : "memory");

    const int NT = NN / JT;   // 16 tiles
    for (int tt = 0; tt < NT; ++tt) {
        const float* __restrict__ csv = (tt & 1) ? cs[1] : cs[0];
        if (tt + 1 < NT) {
            const uint32_t nxt = (tt & 1) ? lds0 : lds1;
            const uint32_t goff = (uint32_t)((tt + 1) * JT * DD * 4) + lchunk;
            asm volatile(
                "global_load_async_to_lds_b128 %0, %2, %4\n\t"
                "global_load_async_to_lds_b128 %1, %3, %4\n\t"
                "s_wait_asynccnt 0x2"                    // tile tt landed
                :: "v"(nxt + lchunk), "v"(nxt + lchunk + 4096),
                   "v"(goff), "v"(goff + 4096), "s"(cbase)
                : "memory");
        } else {
            asm volatile("s_wait_asynccnt 0x0" ::: "memory");
        }
        if (t < JT) mf[t] = mB[tt * JT + t] ? 1.0f : 0.0f;
        __syncthreads();                 // all waves' copies + mf visible
#pragma unroll
        for (int jj = 0; jj < JT; ++jj) {
            const float2 c2 = *(const float2*)(csv + jj * DD + mp);
            const float mv = mf[jj];
            accx = fmaf(mv, fmaxf(av.x + c2.x, 0.0f), accx);
            accy = fmaf(mv, fmaxf(av.y + c2.y, 0.0f), accy);
        }
        __syncthreads();                 // done reading buf before overwrite
    }

    const float rowm = mB[n] ? 1.0f : 0.0f;   // row-side mask
    accx *= rowm;
    accy *= rowm;

    aggS[nl * DD + mp]     = accx;
    aggS[nl * DD + mp + 1] = accy;
    const float2 v2 = *(const float2*)(V + bn * DD + mp);
    vS[nl * DD + mp]     = v2.x;
    vS[nl * DD + mp + 1] = v2.y;
    __syncthreads();

    // out[b,n,o] = relu( sum_k v[n,k]*Wn[o,k] + agg[n,k]*Wn[o,64+k] + b_node[o] )
    float ox = bnode[mp], oy = bnode[mp + 1];
#pragma unroll
    for (int k = 0; k < DD; ++k) {
        const float hv = vS[nl * DD + k];
        ox = fmaf(hv, WnS[mp * 129 + k], ox);
        oy = fmaf(hv, WnS[(mp + 1) * 129 + k], oy);
    }
#pragma unroll
    for (int k = 0; k < DD; ++k) {
        const float ga = aggS[nl * DD + k];
        ox = fmaf(ga, WnS[mp * 129 + 64 + k], ox);
        oy = fmaf(ga, WnS[(mp + 1) * 129 + 64 + k], oy);
    }
    out[bn * DD + mp]     = fmaxf(ox, 0.0f);
    out[bn * DD + mp + 1] = fmaxf(oy, 0.0f);
}

extern "C" void kernel_launch(void* const* d_in, const int* in_sizes, int n_in,
                              void* d_out, int out_size, void* d_ws, size_t ws_size,
                              hipStream_t stream) {
    const float*         V     = (const float*)d_in[0];
    const unsigned char* mask  = (const unsigned char*)d_in[1];  // jnp bool_ = 1B
    const float*         Wmsg  = (const float*)d_in[2];
    const float*         bmsg  = (const float*)d_in[3];
    const float*         Wnode = (const float*)d_in[4];
    const float*         bnode = (const float*)d_in[5];
    float* out = (float*)d_out;

    // workspace: a+b_msg (0.5 MB) followed by c (0.5 MB)
    float* aplus = (float*)d_ws;
    float* cbuf  = aplus + (size_t)NB * NN * DD;

    proj_wmma_kernel<<<1024, 32, 0, stream>>>(V, Wmsg, bmsg, aplus, cbuf);
    agg_mlp_kernel<<<NB * (NN / NTILE), 256, 0, stream>>>(V, mask, aplus, cbuf,
                                                          Wnode, bnode, out);
}